// LESPLoss_73014444032083
// MI455X (gfx1250) — compile-verified
//
#include <hip/hip_runtime.h>
#include <math.h>

typedef float v2f __attribute__((ext_vector_type(2)));
typedef float v8f __attribute__((ext_vector_type(8)));

#define B_DIM 2048
#define C_DIM 1000
#define P_DIM 20
#define ROWS_PER_BLOCK 16

// ---------------------------------------------------------------------------
// Zero the double accumulator in workspace (d_ws is poisoned, must re-init).
// ---------------------------------------------------------------------------
__global__ void lesp_init(double* __restrict__ accum) {
    accum[0] = 0.0;
}

// ---------------------------------------------------------------------------
// One wave (32 lanes) per 16-row tile.
//  Phase 1: S_b = sum_j exp(-x[b,j]) via V_WMMA_F32_16X16X4_F32 with B=ones.
//           A tile = 16x4 exp(-x) values; per-lane A layout (ISA 7.12.2):
//             lanes 0-15  : row m=lane,     K = {0,1} in {VGPR0,VGPR1}
//             lanes 16-31 : row m=lane-16,  K = {2,3}
//           -> lane loads float2 at column (k0 + 2*(lane>>4)); 8B aligned
//              since row stride is 4000 B and the column offset is 8B-even.
//           D layout: lane 0 holds rows 0..7 in acc[0..7]; lane 16 rows 8..15.
//  Phase 2: T_b = sum_{valid p} exp(x[b, t[b,p]]), V_b = #valid, split so
//           lane m does p=0..9 and lane m+16 does p=10..19 for row m.
//  Phase 3: per-row contribution T*S - V, reduce 16 rows, one f64 atomic.
// ---------------------------------------------------------------------------
__global__ __launch_bounds__(32) void lesp_rows(const float* __restrict__ x,
                                                const int*   __restrict__ tgt,
                                                double*      __restrict__ accum) {
    __shared__ float sS[ROWS_PER_BLOCK];
    __shared__ float sT[32];
    __shared__ int   sV[32];
    __shared__ float sC[ROWS_PER_BLOCK];

    const int lane = threadIdx.x;        // 0..31 (wave32)
    const int half = lane >> 4;          // 0 or 1
    const int m    = lane & 15;          // row within the 16-row tile
    const int row  = blockIdx.x * ROWS_PER_BLOCK + m;

    const float* xrow = x + (size_t)row * C_DIM;

    // ---- Phase 1: row sums of exp(-x) via FP32 WMMA ----
    v8f acc = {};                        // C/D accumulator, starts at zero
    v2f bones;                           // B = all ones -> D[m][n] = row sum
    bones[0] = 1.0f;
    bones[1] = 1.0f;

    const int cbase = 2 * half;          // this lane's column pair within tile
    for (int k0 = 0; k0 < C_DIM; k0 += 4) {          // 250 K-tiles, no tail
        const v2f xv = *(const v2f*)(xrow + k0 + cbase);
        v2f a;
        a[0] = __expf(-xv[0]);
        a[1] = __expf(-xv[1]);
        acc = __builtin_amdgcn_wmma_f32_16x16x4_f32(
                  /*neg_a=*/false, a, /*neg_b=*/false, bones,
                  /*c_mod=*/(short)0, acc, /*reuse_a=*/false, /*reuse_b=*/false);
    }

    // Extract row sums: lane 0 -> rows 0..7, lane 16 -> rows 8..15 (col n=0).
    if (lane == 0) {
#pragma unroll
        for (int r = 0; r < 8; ++r) sS[r] = acc[r];
    }
    if (lane == 16) {
#pragma unroll
        for (int r = 0; r < 8; ++r) sS[8 + r] = acc[r];
    }

    // ---- Phase 2: gather term over targets (P=20 split 10/10 per lane pair) ----
    float tpart = 0.0f;
    int   vcnt  = 0;
    const int* trow = tgt + row * P_DIM;
#pragma unroll
    for (int i = 0; i < 10; ++i) {
        const int p = half * 10 + i;
        const int t = trow[p];
        if (t > -1) {
            tpart += __expf(xrow[t]);
            vcnt  += 1;
        }
    }
    sT[lane] = tpart;
    sV[lane] = vcnt;
    __syncthreads();

    // ---- Phase 3: per-row contribution, block reduction, one f64 atomic ----
    if (lane < ROWS_PER_BLOCK) {
        const float T = sT[lane] + sT[lane + 16];
        const int   V = sV[lane] + sV[lane + 16];
        const float S = sS[lane];
        sC[lane] = T * S - (float)V;     // exp(x_t)*exp(-x_t) == exp(0) == 1
    }
    __syncthreads();

    if (lane == 0) {
        double s = 0.0;
#pragma unroll
        for (int r = 0; r < ROWS_PER_BLOCK; ++r) s += (double)sC[r];
        atomicAdd(accum, s);             // global_atomic_add_f64
    }
}

// ---------------------------------------------------------------------------
// loss = log(1 + loss_data) / C
// ---------------------------------------------------------------------------
__global__ void lesp_finalize(const double* __restrict__ accum,
                              float* __restrict__ out) {
    out[0] = (float)(log(1.0 + accum[0]) / (double)C_DIM);
}

extern "C" void kernel_launch(void* const* d_in, const int* in_sizes, int n_in,
                              void* d_out, int out_size, void* d_ws, size_t ws_size,
                              hipStream_t stream) {
    const float* x   = (const float*)d_in[0];   // [B, C] fp32
    const int*   tgt = (const int*)d_in[1];     // [B, P] int32
    float*  out   = (float*)d_out;              // scalar fp32
    double* accum = (double*)d_ws;              // 8-byte scratch accumulator

    lesp_init<<<1, 1, 0, stream>>>(accum);
    lesp_rows<<<B_DIM / ROWS_PER_BLOCK, 32, 0, stream>>>(x, tgt, accum);
    lesp_finalize<<<1, 1, 0, stream>>>(accum, out);
}